// ExpertBuffer_24833500906107
// MI455X (gfx1250) — compile-verified
//
#include <hip/hip_runtime.h>
#include <hip/hip_bf16.h>
#include <stdint.h>

// ExpertBuffer fetch_on_demand: pure gather/scatter copy (~400 MB traffic,
// ~17 us roofline at 23.3 TB/s). Bulk data is moved with the CDNA5 async
// global<->LDS DMA path (ASYNCcnt-tracked, 16 B/lane, no VGPR staging).

#define EB_BLOCK  256
#define EB_UNROLL 8
#define EB_TILE   (EB_BLOCK * EB_UNROLL)   // float4s per block tile = 2048 (32 KiB)

// out[slot] = (slot appears in slot_ids) ? src[expert_ids[last match]] : cache[slot]
__global__ void __launch_bounds__(EB_BLOCK)
eb_gather_rows_async(float4* __restrict__ dst,
                     const float4* __restrict__ src,
                     const float4* __restrict__ cache,
                     const int* __restrict__ expert_ids,
                     const int* __restrict__ slot_ids,
                     int n_pairs,
                     long row_vec4)     // row length in float4 units
{
    const int slot = blockIdx.y;

    // Resolve which expert (if any) maps to this cache slot; last pair wins,
    // matching sequential scatter semantics.
    int e = -1;
    for (int k = 0; k < n_pairs; ++k) {
        if (slot_ids[k] == slot) e = expert_ids[k];
    }
    const float4* srow = (e >= 0) ? (src + (long)e * row_vec4)
                                  : (cache + (long)slot * row_vec4);
    float4* drow = dst + (long)slot * row_vec4;

    const long base = (long)blockIdx.x * EB_TILE + (long)threadIdx.x;

    __shared__ float4 buf[EB_TILE];
    // Low 32 bits of an LDS flat address == group-segment byte offset.
    const uint32_t lds0 = (uint32_t)(size_t)(void*)buf;

    // Stage: global -> LDS via async DMA (no VGPR data path).
#pragma unroll
    for (int u = 0; u < EB_UNROLL; ++u) {
        const long i = base + (long)u * EB_BLOCK;
        if (i < row_vec4) {
            const float4* gp = srow + i;
            const uint32_t lp = lds0 + (uint32_t)(threadIdx.x + u * EB_BLOCK) * 16u;
            asm volatile("global_load_async_to_lds_b128 %0, %1, off"
                         :: "v"(lp), "v"(gp) : "memory");
        }
    }
    asm volatile("s_wait_asynccnt 0x0" ::: "memory");

    // Drain: LDS -> global via async DMA.
#pragma unroll
    for (int u = 0; u < EB_UNROLL; ++u) {
        const long i = base + (long)u * EB_BLOCK;
        if (i < row_vec4) {
            float4* gp = drow + i;
            const uint32_t lp = lds0 + (uint32_t)(threadIdx.x + u * EB_BLOCK) * 16u;
            asm volatile("global_store_async_from_lds_b128 %0, %1, off"
                         :: "v"(gp), "v"(lp) : "memory");
        }
    }
    asm volatile("s_wait_asynccnt 0x0" ::: "memory");
}

static inline void launch_copy(float* dst, const float* src, const float* cache,
                               const int* eid, const int* sid, int n_pairs,
                               long row_elems, int n_slots, hipStream_t stream)
{
    const long row_vec4 = row_elems / 4;   // all rows are multiples of 4 floats
    const int  gx = (int)((row_vec4 + EB_TILE - 1) / EB_TILE);
    dim3 grid((unsigned)gx, (unsigned)n_slots, 1);
    eb_gather_rows_async<<<grid, EB_BLOCK, 0, stream>>>(
        (float4*)dst, (const float4*)src, (const float4*)cache,
        eid, sid, n_pairs, row_vec4);
}

extern "C" void kernel_launch(void* const* d_in, const int* in_sizes, int n_in,
                              void* d_out, int out_size, void* d_ws, size_t ws_size,
                              hipStream_t stream)
{
    const float* w13_src    = (const float*)d_in[0];
    const float* w13b_src   = (const float*)d_in[1];
    const float* w2_src     = (const float*)d_in[2];
    const float* w2b_src    = (const float*)d_in[3];
    const float* w13_cache  = (const float*)d_in[4];
    const float* w13b_cache = (const float*)d_in[5];
    const float* w2_cache   = (const float*)d_in[6];
    const float* w2b_cache  = (const float*)d_in[7];
    const int*   eid        = (const int*)d_in[8];
    const int*   sid        = (const int*)d_in[9];

    const int n_pairs = in_sizes[8];
    const int n_slots = in_sizes[9];

    const long R13  = 2L * 2048L * 1024L;  // 4,194,304 elems (16 MiB/slot)
    const long R13B = 2L * 2048L;          // 4,096
    const long R2   = 1024L * 2048L;       // 2,097,152 (8 MiB/slot)
    const long R2B  = 1024L;

    float* out    = (float*)d_out;
    float* o_w13  = out;
    float* o_w13b = o_w13  + (long)n_slots * R13;
    float* o_w2   = o_w13b + (long)n_slots * R13B;
    float* o_w2b  = o_w2   + (long)n_slots * R2;

    // Big tensors first (dominate traffic), then biases.
    launch_copy(o_w13,  w13_src,  w13_cache,  eid, sid, n_pairs, R13,  n_slots, stream);
    launch_copy(o_w2,   w2_src,   w2_cache,   eid, sid, n_pairs, R2,   n_slots, stream);
    launch_copy(o_w13b, w13b_src, w13b_cache, eid, sid, n_pairs, R13B, n_slots, stream);
    launch_copy(o_w2b,  w2b_src,  w2b_cache,  eid, sid, n_pairs, R2B,  n_slots, stream);
}